// GCN_67817533604372
// MI455X (gfx1250) — compile-verified
//
#include <hip/hip_runtime.h>
#include <hip/hip_bf16.h>
#include <stdint.h>

typedef float v2f __attribute__((ext_vector_type(2)));
typedef float v8f __attribute__((ext_vector_type(8)));
typedef uint32_t u32x4 __attribute__((ext_vector_type(4)));
typedef uint32_t u32x8 __attribute__((ext_vector_type(8)));

static constexpr int N_NODES = 100000;
static constexpr int N_EDGES = 800000;
static constexpr int F_IN    = 128;
static constexpr int F_HID   = 128;
static constexpr int N_CLS   = 64;

// ---------------------------------------------------------------------------
// Degree / normalization
// ---------------------------------------------------------------------------
__global__ void k_init_deg(float* __restrict__ deg, int n) {
    int i = blockIdx.x * blockDim.x + threadIdx.x;
    if (i < n) deg[i] = 1.0f;  // self-loop contributes 1 to every in-degree
}

__global__ void k_count_deg(const int* __restrict__ dst, float* __restrict__ deg, int e) {
    int i = blockIdx.x * blockDim.x + threadIdx.x;
    if (i < e) atomicAdd(&deg[dst[i]], 1.0f);
}

__global__ void k_deg_to_dis(float* __restrict__ deg, int n) {
    int i = blockIdx.x * blockDim.x + threadIdx.x;
    if (i < n) deg[i] = rsqrtf(deg[i]);  // deg >= 1 always (self loops)
}

// ---------------------------------------------------------------------------
// FP32 WMMA GEMM: D[rows, ncols] = A[rows, K] @ B[K, ncols]
// grid.x = rows/16 (one block per 16-row tile); blockDim = (ncols/16)*32,
// one wave per 16x16 output tile. The shared 16xK A tile is staged into LDS
// once per block by the Tensor Data Mover (1-D D# descriptor, TENSORcnt),
// removing the duplicated A reads from L2.
//
// ISA 7.12.2 fp32 operand layouts (wave32):
//   A (16x4, MxK):  vgpr = K%2, lane = (K/2)*16 + M
//   B (4x16, KxN):  vgpr = K%2, lane = (K/2)*16 + N
//   D (16x16):      vgpr = M%8, lane = (M/8)*16 + N
// ---------------------------------------------------------------------------
__global__ void k_gemm_wmma_f32(const float* __restrict__ A,
                                const float* __restrict__ B,
                                float* __restrict__ D,
                                int K, int ncols) {
    __shared__ float sA[16 * 128];           // staged A tile (8KB @ K=128)

    const int rowTile = blockIdx.x;
    const int wave    = threadIdx.x >> 5;
    const int lane    = threadIdx.x & 31;
    const int row0    = rowTile * 16;
    const int col0    = wave * 16;

    // ---- wave 0: TDM 1-D copy of the 16xK fp32 A tile into LDS ----
    if (threadIdx.x < 32) {                  // wave-uniform branch
        void*    sAp   = (void*)sA;                    // escapes into the asm below
        uint32_t ldsad = (uint32_t)(uintptr_t)sAp;     // low 32 flat bits = LDS byte addr
        uint64_t gaddr = (uint64_t)(const void*)(A + (size_t)row0 * K);
        uint32_t nelem = 16u * (uint32_t)K;            // 4-byte elements (2048 @ K=128)

        u32x4 g0;
        g0[0] = 1u;                                    // count=1, user D#, gather off
        g0[1] = ldsad;                                 // lds_addr (bytes)
        g0[2] = (uint32_t)gaddr;                       // global_addr[31:0]
        g0[3] = (uint32_t)(gaddr >> 32) | (2u << 30);  // global_addr[56:32] | type=2

        u32x8 g1;
        g1[0] = 2u << 16;                              // data_size=2 -> 4-byte elements
        g1[1] = (nelem & 0xFFFFu) << 16;               // tensor_dim0[15:0]  (bits 63:48)
        g1[2] = (nelem >> 16) | (1u << 16);            // tensor_dim0[31:16] | tensor_dim1=1
        g1[3] = (nelem & 0xFFFFu) << 16;               // tensor_dim1[31:16]=0 | tile_dim0
        g1[4] = 1u;                                    // tile_dim1=1, tile_dim2=0
        g1[5] = nelem;                                 // tensor_dim0_stride[31:0]
        g1[6] = 0u;                                    // stride0[47:32] | dim1_stride[15:0]
        g1[7] = 0u;

        // %2 (the LDS pointer) is unused in the template but makes sA escape so
        // the "memory" clobber covers the TDM's LDS writes (prevents the
        // compiler from folding the subsequent ds_loads of sA to undef).
        asm volatile("tensor_load_to_lds %0, %1"
                     :: "s"(g0), "s"(g1), "r"(sAp)
                     : "memory");
        __builtin_amdgcn_s_wait_tensorcnt(0);
    }
    __syncthreads();

    const int half = lane >> 4;              // 16-lane half
    const int lid  = lane & 15;              // M (A) or N (B/D) within tile

    const float* sArow = sA + lid * K + 2 * half;                    // K = 2h, 2h+1
    const float* bcol  = B + (size_t)(2 * half) * ncols + col0 + lid;

    v8f acc = {};
#pragma unroll 8
    for (int k = 0; k < K; k += 4) {
        v2f a = *(const v2f*)(sArow + k);            // ds_load_b64 (8B aligned)
        v2f b;
        b.x = bcol[(size_t)k * ncols];               // row k + 2*half
        b.y = bcol[(size_t)k * ncols + ncols];       // row k + 2*half + 1
        acc = __builtin_amdgcn_wmma_f32_16x16x4_f32(
            /*neg_a=*/false, a, /*neg_b=*/false, b,
            /*c_mod=*/(short)0, acc, /*reuse_a=*/false, /*reuse_b=*/false);
    }

    float* drow = D + (size_t)(row0 + 8 * half) * ncols + col0 + lid;
#pragma unroll
    for (int g = 0; g < 8; ++g)
        drow[(size_t)g * ncols] = acc[g];
}

// ---------------------------------------------------------------------------
// Aggregation: acc[n] = h[n]*dis[n]^2 (self loop), then edge scatter-add.
// Vectorized float4: one thread per (edge, 4-feature quad) -> b128 load + 4
// global_atomic_add_f32 into the L2-resident accumulator.
// ---------------------------------------------------------------------------
__global__ void k_init_acc4(const float4* __restrict__ h4, const float* __restrict__ dis,
                            float4* __restrict__ acc4, int n, int qshift) {
    int t = blockIdx.x * blockDim.x + threadIdx.x;
    int total = n << qshift;
    if (t >= total) return;
    float dd = dis[t >> qshift];
    dd *= dd;
    float4 v = h4[t];
    acc4[t] = make_float4(v.x * dd, v.y * dd, v.z * dd, v.w * dd);
}

__global__ void k_edge_scatter4(const float4* __restrict__ h4, const float* __restrict__ dis,
                                const int* __restrict__ src, const int* __restrict__ dst,
                                float* __restrict__ acc, int e, int qshift) {
    int t = blockIdx.x * blockDim.x + threadIdx.x;
    int total = e << qshift;     // <= 25.6M, fits int
    if (t >= total) return;
    int edge = t >> qshift;
    int q    = t & ((1 << qshift) - 1);
    int s = src[edge];
    int d = dst[edge];
    float norm = dis[s] * dis[d];
    float4 v = h4[((size_t)s << qshift) + q];
    float* out = acc + ((((size_t)d << qshift) + q) << 2);
    atomicAdd(out + 0, v.x * norm);
    atomicAdd(out + 1, v.y * norm);
    atomicAdd(out + 2, v.z * norm);
    atomicAdd(out + 3, v.w * norm);
}

// layer1 epilogue: +b1, write embeddings, write relu(h) for layer 2
__global__ void k_finish1(const float4* __restrict__ acc4, const float4* __restrict__ b4,
                          float4* __restrict__ emb4, float4* __restrict__ hrelu4, int total) {
    int t = blockIdx.x * blockDim.x + threadIdx.x;
    if (t >= total) return;
    float4 a = acc4[t];
    float4 b = b4[t & ((F_HID / 4) - 1)];    // 32 quads per 128-wide row
    float4 v = make_float4(a.x + b.x, a.y + b.y, a.z + b.z, a.w + b.w);
    emb4[t]   = v;
    hrelu4[t] = make_float4(fmaxf(v.x, 0.f), fmaxf(v.y, 0.f),
                            fmaxf(v.z, 0.f), fmaxf(v.w, 0.f));
}

// layer2 epilogue: +b2 then row-wise log_softmax over 64 classes (wave per row)
__global__ void k_logsoftmax(const float* __restrict__ acc, const float* __restrict__ b,
                             float* __restrict__ out, int n) {
    int wave = (int)((blockIdx.x * blockDim.x + threadIdx.x) >> 5);
    int lane = threadIdx.x & 31;
    if (wave >= n) return;
    const float* row = acc + (size_t)wave * N_CLS;
    float v0 = row[lane]      + b[lane];
    float v1 = row[lane + 32] + b[lane + 32];

    float m = fmaxf(v0, v1);
#pragma unroll
    for (int o = 16; o > 0; o >>= 1) m = fmaxf(m, __shfl_xor(m, o, 32));
    float s = expf(v0 - m) + expf(v1 - m);
#pragma unroll
    for (int o = 16; o > 0; o >>= 1) s += __shfl_xor(s, o, 32);
    float ls = logf(s);

    float* orow = out + (size_t)wave * N_CLS;
    orow[lane]      = v0 - m - ls;
    orow[lane + 32] = v1 - m - ls;
}

// ---------------------------------------------------------------------------
// Launcher
// ---------------------------------------------------------------------------
extern "C" void kernel_launch(void* const* d_in, const int* in_sizes, int n_in,
                              void* d_out, int out_size, void* d_ws, size_t ws_size,
                              hipStream_t stream) {
    const float* x  = (const float*)d_in[0];   // [N, 128]
    const int*   ei = (const int*)d_in[1];     // [2, E]
    const float* W1 = (const float*)d_in[2];   // [128, 128]
    const float* b1 = (const float*)d_in[3];   // [128]
    const float* W2 = (const float*)d_in[4];   // [128, 64]
    const float* b2 = (const float*)d_in[5];   // [64]
    const int* src = ei;
    const int* dst = ei + N_EDGES;

    float* outLS  = (float*)d_out;                       // [N, 64] log_softmax
    float* outEmb = outLS + (size_t)N_NODES * N_CLS;     // [N, 128] embeddings

    // Workspace (floats), 256B-aligned regions:
    //   dis : N        (deg -> dis in place)
    //   h1  : N*128    (x@W1, later reused for relu(h))
    //   buf : N*128    (acc1; later split: [0,N*64)=h2, [N*64,N*128)=acc2)
    float* dis  = (float*)d_ws;
    float* h1   = dis + 100096;                          // pad 100000->100096 (x4 = 256B mult)
    float* buf  = h1 + (size_t)N_NODES * F_HID;
    float* acc1 = buf;
    float* h2   = buf;
    float* acc2 = buf + (size_t)N_NODES * N_CLS;

    const int B = 256;
    auto blocks = [](long long total, int b) { return (int)((total + b - 1) / b); };

    // --- normalization coefficients ---
    k_init_deg  <<<blocks(N_NODES, B), B, 0, stream>>>(dis, N_NODES);
    k_count_deg <<<blocks(N_EDGES, B), B, 0, stream>>>(dst, dis, N_EDGES);
    k_deg_to_dis<<<blocks(N_NODES, B), B, 0, stream>>>(dis, N_NODES);

    // --- layer 1: h1 = x @ W1 (TDM-staged WMMA), aggregate, +b1, emb + relu ---
    k_gemm_wmma_f32<<<N_NODES / 16, (F_HID / 16) * 32, 0, stream>>>(x, W1, h1, F_IN, F_HID);
    k_init_acc4    <<<blocks((long long)N_NODES << 5, B), B, 0, stream>>>(
        (const float4*)h1, dis, (float4*)acc1, N_NODES, 5);
    k_edge_scatter4<<<blocks((long long)N_EDGES << 5, B), B, 0, stream>>>(
        (const float4*)h1, dis, src, dst, acc1, N_EDGES, 5);
    k_finish1      <<<blocks((long long)N_NODES << 5, B), B, 0, stream>>>(
        (const float4*)acc1, (const float4*)b1, (float4*)outEmb, (float4*)h1,
        N_NODES * (F_HID / 4));

    // --- layer 2: h2 = relu(h1) @ W2 (WMMA), aggregate, +b2, log_softmax ---
    k_gemm_wmma_f32<<<N_NODES / 16, (N_CLS / 16) * 32, 0, stream>>>(h1, W2, h2, F_HID, N_CLS);
    k_init_acc4    <<<blocks((long long)N_NODES << 4, B), B, 0, stream>>>(
        (const float4*)h2, dis, (float4*)acc2, N_NODES, 4);
    k_edge_scatter4<<<blocks((long long)N_EDGES << 4, B), B, 0, stream>>>(
        (const float4*)h2, dis, src, dst, acc2, N_EDGES, 4);
    k_logsoftmax   <<<blocks((long long)N_NODES * 32, B), B, 0, stream>>>(
        acc2, b2, outLS, N_NODES);
}